// GAT_23828478558586
// MI455X (gfx1250) — compile-verified
//
#include <hip/hip_runtime.h>
#include <math.h>

// Problem constants (match reference)
#define N_NODES 50000
#define N_EDGES 800000
#define F_IN    512
#define HID     32
#define HEADS   8
#define CLS     64
#define NEG_SLOPE 0.2f
#define EPS_SM  1e-16f

typedef __attribute__((ext_vector_type(2))) float v2f;
typedef __attribute__((ext_vector_type(4))) float v4f;
typedef __attribute__((ext_vector_type(8))) float v8f;

__device__ __forceinline__ float lrelu(float x) { return x > 0.f ? x : NEG_SLOPE * x; }

// float atomic max via sign-magnitude ordering trick (works with -inf init)
__device__ __forceinline__ void atomicMaxF(float* addr, float v) {
    if (v >= 0.f) atomicMax((int*)addr, __float_as_int(v));
    else          atomicMin((unsigned int*)addr, __float_as_uint(v));
}

// ---------------------------------------------------------------- fill
__global__ void k_fill(float* __restrict__ p, float val, int n) {
    int i = blockIdx.x * blockDim.x + threadIdx.x;
    int stride = gridDim.x * blockDim.x;
    for (; i < n; i += stride) p[i] = val;
}

// ---------------------------------------------------------------- GEMM via fp32 WMMA
// C[M,Nw] = A[M,K] @ B[K,Nw].  Block tile 64x64, 8 waves (4 along M, 2 along N),
// each wave owns two 16x16 WMMA accumulators.  K multiple of 16, Nw multiple of 64,
// M ragged (guarded).  Global->LDS staging is register double-buffered so the
// next K-chunk's global_load_b128s are in flight during the WMMA phase.
#define BM 64
#define BN 64
#define BK 16

__global__ __launch_bounds__(256) void k_gemm_wmma(const float* __restrict__ A,
                                                   const float* __restrict__ B,
                                                   float* __restrict__ C,
                                                   int M, int K, int Nw) {
    __shared__ float As[BM][BK + 4];   // +4 pad: conflict-free, keeps 16B alignment
    __shared__ float Bs[BK][BN];

    const int tid  = threadIdx.x;
    const int lane = tid & 31;
    const int wave = tid >> 5;
    const int wm   = wave & 3;        // 0..3  (M direction, 16 rows each)
    const int wn   = wave >> 2;       // 0..1  (N direction, 32 cols each)
    const int m0   = blockIdx.y * BM;
    const int n0   = blockIdx.x * BN;
    const int lm   = lane & 15;       // M (for A) / N (for B,C) index within tile
    const int kh   = lane >> 4;       // which K-half the lane holds

    const int arow = tid >> 2,  acol = (tid & 3) << 2;   // A tile: 64 rows x 4 float4
    const int brow = tid >> 4,  bcol = (tid & 15) << 2;  // B tile: 16 rows x 16 float4
    const int gr   = m0 + arow;
    const bool arow_ok = (gr < M);

    v8f acc0 = {};
    v8f acc1 = {};

    // prologue: fetch chunk 0 into registers
    v4f av = {};
    if (arow_ok) av = *(const v4f*)(A + (size_t)gr * K + acol);
    v4f bv = *(const v4f*)(B + (size_t)brow * Nw + n0 + bcol);

    for (int kk = 0; kk < K; kk += BK) {
        // commit current chunk to LDS
        *(v4f*)&As[arow][acol] = av;
        *(v4f*)&Bs[brow][bcol] = bv;
        __syncthreads();

        // issue next chunk's global loads now; they complete during compute
        if (kk + BK < K) {
            av = (v4f){};
            if (arow_ok) av = *(const v4f*)(A + (size_t)gr * K + (kk + BK) + acol);
            bv = *(const v4f*)(B + (size_t)(kk + BK + brow) * Nw + n0 + bcol);
        }

        #pragma unroll
        for (int kb = 0; kb < 4; ++kb) {
            const int k0 = kb * 4 + kh * 2;   // this lane's K pair per ISA A/B layout
            v2f a;
            a.x = As[wm * 16 + lm][k0];
            a.y = As[wm * 16 + lm][k0 + 1];
            v2f b0, b1;
            b0.x = Bs[k0][wn * 32 + lm];
            b0.y = Bs[k0 + 1][wn * 32 + lm];
            b1.x = Bs[k0][wn * 32 + 16 + lm];
            b1.y = Bs[k0 + 1][wn * 32 + 16 + lm];
            acc0 = __builtin_amdgcn_wmma_f32_16x16x4_f32(false, a, false, b0,
                                                         (short)0, acc0, false, false);
            acc1 = __builtin_amdgcn_wmma_f32_16x16x4_f32(false, a, false, b1,
                                                         (short)0, acc1, false, false);
        }
        __syncthreads();
    }

    // C/D layout: VGPR g -> rows g (lanes 0-15) and g+8 (lanes 16-31), col = lane&15
    #pragma unroll
    for (int g = 0; g < 8; ++g) {
        int row = m0 + wm * 16 + g + 8 * kh;
        if (row < M) {
            int col = n0 + wn * 32 + lm;
            C[(size_t)row * Nw + col]      = acc0[g];
            C[(size_t)row * Nw + col + 16] = acc1[g];
        }
    }
}

// ---------------------------------------------------------------- attention scores
// a_src[n,h] = dot(h[n,h,:], att_src[h,:]), same for dst.  One wave per node.
// Requires H*C % 32 == 0 and C % 32 == 0 (true for 8x32 and 1x64).
template<int H, int C>
__global__ void k_att(const float* __restrict__ hfeat,
                      const float* __restrict__ att_s,
                      const float* __restrict__ att_d,
                      float* __restrict__ a_s, float* __restrict__ a_d,
                      int n_nodes) {
    const int HC = H * C, J = HC / 32;
    const int lane = threadIdx.x & 31;
    int gw = (blockIdx.x * blockDim.x + threadIdx.x) >> 5;
    int nw = (gridDim.x * blockDim.x) >> 5;
    for (int n = gw; n < n_nodes; n += nw) {
        float ps[H], pd[H];
        #pragma unroll
        for (int i = 0; i < H; ++i) { ps[i] = 0.f; pd[i] = 0.f; }
        #pragma unroll
        for (int j = 0; j < J; ++j) {
            const int hd = (j * 32) / C;        // C%32==0 -> head is lane-invariant
            int e = j * 32 + lane;
            float v = hfeat[(size_t)n * HC + e];
            ps[hd] += v * att_s[e];
            pd[hd] += v * att_d[e];
        }
        #pragma unroll
        for (int i = 0; i < H; ++i) {
            float a = ps[i], b = pd[i];
            #pragma unroll
            for (int off = 16; off; off >>= 1) {
                a += __shfl_xor(a, off);
                b += __shfl_xor(b, off);
            }
            if (lane == 0) {
                a_s[(size_t)n * H + i] = a;
                a_d[(size_t)n * H + i] = b;
            }
        }
    }
}

// ---------------------------------------------------------------- segment max / sum
__global__ void k_edge_max(const int* __restrict__ ei,
                           const float* __restrict__ a_s, const float* __restrict__ a_d,
                           float* __restrict__ m, int H) {
    const int ET = N_EDGES + N_NODES;
    int i = blockIdx.x * blockDim.x + threadIdx.x;
    int stride = gridDim.x * blockDim.x;
    for (; i < ET; i += stride) {
        int s = (i < N_EDGES) ? ei[i]           : (i - N_EDGES);
        int d = (i < N_EDGES) ? ei[N_EDGES + i] : (i - N_EDGES);
        for (int hh = 0; hh < H; ++hh) {
            float v = lrelu(a_s[(size_t)s * H + hh] + a_d[(size_t)d * H + hh]);
            atomicMaxF(&m[(size_t)d * H + hh], v);
        }
    }
}

__global__ void k_edge_sum(const int* __restrict__ ei,
                           const float* __restrict__ a_s, const float* __restrict__ a_d,
                           const float* __restrict__ m, float* __restrict__ ssum, int H) {
    const int ET = N_EDGES + N_NODES;
    int i = blockIdx.x * blockDim.x + threadIdx.x;
    int stride = gridDim.x * blockDim.x;
    for (; i < ET; i += stride) {
        int s = (i < N_EDGES) ? ei[i]           : (i - N_EDGES);
        int d = (i < N_EDGES) ? ei[N_EDGES + i] : (i - N_EDGES);
        for (int hh = 0; hh < H; ++hh) {
            float v = lrelu(a_s[(size_t)s * H + hh] + a_d[(size_t)d * H + hh]);
            atomicAdd(&ssum[(size_t)d * H + hh], expf(v - m[(size_t)d * H + hh]));
        }
    }
}

// ---------------------------------------------------------------- weighted scatter-add
// One wave per edge: 1KB contiguous gather of h[src], atomic scatter to out[dst].
template<int H, int C>
__global__ void k_edge_agg(const int* __restrict__ ei,
                           const float* __restrict__ hfeat,
                           const float* __restrict__ a_s, const float* __restrict__ a_d,
                           const float* __restrict__ m, const float* __restrict__ ssum,
                           float* __restrict__ out) {
    const int HC = H * C, J = HC / 32;
    const int ET = N_EDGES + N_NODES;
    const int lane = threadIdx.x & 31;
    int gw = (blockIdx.x * blockDim.x + threadIdx.x) >> 5;
    int nw = (gridDim.x * blockDim.x) >> 5;
    for (int e = gw; e < ET; e += nw) {
        int s = (e < N_EDGES) ? ei[e]           : (e - N_EDGES);
        int d = (e < N_EDGES) ? ei[N_EDGES + e] : (e - N_EDGES);
        float al = 0.f;
        if (lane < H) {
            int idx = d * H + lane;
            float v = lrelu(a_s[s * H + lane] + a_d[idx]);
            al = expf(v - m[idx]) / (ssum[idx] + EPS_SM);
        }
        #pragma unroll
        for (int j = 0; j < J; ++j) {
            const int hd = (j * 32) / C;
            float a = __shfl(al, hd);
            int el = j * 32 + lane;
            atomicAdd(&out[(size_t)d * HC + el], hfeat[(size_t)s * HC + el] * a);
        }
    }
}

// ---------------------------------------------------------------- bias + relu (in place)
__global__ void k_bias_relu(float* __restrict__ x, const float* __restrict__ b,
                            int n, int F) {
    int i = blockIdx.x * blockDim.x + threadIdx.x;
    int stride = gridDim.x * blockDim.x;
    int tot = n * F;
    for (; i < tot; i += stride) x[i] = fmaxf(x[i] + b[i % F], 0.f);
}

// ---------------------------------------------------------------- bias + log_softmax(64)
__global__ void k_logsoftmax(const float* __restrict__ o2, const float* __restrict__ b2,
                             float* __restrict__ out, int n_nodes) {
    const int lane = threadIdx.x & 31;
    int gw = (blockIdx.x * blockDim.x + threadIdx.x) >> 5;
    int nw = (gridDim.x * blockDim.x) >> 5;
    for (int n = gw; n < n_nodes; n += nw) {
        float v0 = o2[(size_t)n * CLS + lane]      + b2[lane];
        float v1 = o2[(size_t)n * CLS + 32 + lane] + b2[32 + lane];
        float mx = fmaxf(v0, v1);
        #pragma unroll
        for (int off = 16; off; off >>= 1) mx = fmaxf(mx, __shfl_xor(mx, off));
        float se = expf(v0 - mx) + expf(v1 - mx);
        #pragma unroll
        for (int off = 16; off; off >>= 1) se += __shfl_xor(se, off);
        float l = logf(se);
        out[(size_t)n * CLS + lane]      = v0 - mx - l;
        out[(size_t)n * CLS + 32 + lane] = v1 - mx - l;
    }
}

// ---------------------------------------------------------------- launch
extern "C" void kernel_launch(void* const* d_in, const int* in_sizes, int n_in,
                              void* d_out, int out_size, void* d_ws, size_t ws_size,
                              hipStream_t stream) {
    const float* x    = (const float*)d_in[0];
    const int*   ei   = (const int*)d_in[1];     // [2,E] row-major (src row, dst row)
    const float* W1   = (const float*)d_in[2];   // [512, 256]
    const float* as1  = (const float*)d_in[3];   // [8,32]
    const float* ad1  = (const float*)d_in[4];
    const float* b1   = (const float*)d_in[5];   // [256]
    const float* W2   = (const float*)d_in[6];   // [256, 64]
    const float* as2  = (const float*)d_in[7];   // [1,64]
    const float* ad2  = (const float*)d_in[8];
    const float* b2   = (const float*)d_in[9];   // [64]
    float* out = (float*)d_out;

    const int N = N_NODES;
    const int HC1 = HEADS * HID;                 // 256

    // workspace layout (floats)
    float* h1    = (float*)d_ws;                 // N*256
    float* out1  = h1    + (size_t)N * HC1;      // N*256
    float* asrc1 = out1  + (size_t)N * HC1;      // N*8
    float* adst1 = asrc1 + (size_t)N * HEADS;
    float* m1    = adst1 + (size_t)N * HEADS;
    float* s1    = m1    + (size_t)N * HEADS;
    float* h2    = s1    + (size_t)N * HEADS;    // N*64
    float* out2  = h2    + (size_t)N * CLS;      // N*64
    float* asrc2 = out2  + (size_t)N * CLS;      // N
    float* adst2 = asrc2 + N;
    float* m2    = adst2 + N;
    float* s2    = m2    + N;

    const float NINF = -INFINITY;
    // re-init accumulators every call (graph-replay safe)
    k_fill<<<2048, 256, 0, stream>>>(out1, 0.f, N * HC1);
    k_fill<<<512,  256, 0, stream>>>(out2, 0.f, N * CLS);
    k_fill<<<256,  256, 0, stream>>>(s1,   0.f, N * HEADS);
    k_fill<<<256,  256, 0, stream>>>(m1,   NINF, N * HEADS);
    k_fill<<<64,   256, 0, stream>>>(s2,   0.f, N);
    k_fill<<<64,   256, 0, stream>>>(m2,   NINF, N);

    // ---- layer 1
    {
        dim3 grid(HC1 / BN, (N + BM - 1) / BM);  // (4, 782)
        k_gemm_wmma<<<grid, 256, 0, stream>>>(x, W1, h1, N, F_IN, HC1);
    }
    k_att<HEADS, HID><<<6250, 256, 0, stream>>>(h1, as1, ad1, asrc1, adst1, N);

    const int ET = N_EDGES + N_NODES;
    int eb = (ET + 255) / 256;
    k_edge_max<<<eb, 256, 0, stream>>>(ei, asrc1, adst1, m1, HEADS);
    k_edge_sum<<<eb, 256, 0, stream>>>(ei, asrc1, adst1, m1, s1, HEADS);
    int wb = (ET * 32 + 255) / 256;  // one wave per edge
    k_edge_agg<HEADS, HID><<<wb, 256, 0, stream>>>(ei, h1, asrc1, adst1, m1, s1, out1);

    k_bias_relu<<<4096, 256, 0, stream>>>(out1, b1, N, HC1);

    // ---- layer 2
    {
        dim3 grid(CLS / BN, (N + BM - 1) / BM);  // (1, 782)
        k_gemm_wmma<<<grid, 256, 0, stream>>>(out1, W2, h2, N, HC1, CLS);
    }
    k_att<1, CLS><<<6250, 256, 0, stream>>>(h2, as2, ad2, asrc2, adst2, N);

    k_edge_max<<<eb, 256, 0, stream>>>(ei, asrc2, adst2, m2, 1);
    k_edge_sum<<<eb, 256, 0, stream>>>(ei, asrc2, adst2, m2, s2, 1);
    k_edge_agg<1, CLS><<<wb, 256, 0, stream>>>(ei, h2, asrc2, adst2, m2, s2, out2);

    k_logsoftmax<<<6250, 256, 0, stream>>>(out2, b2, out, N);
}